// DiDiAttention_62457414419144
// MI455X (gfx1250) — compile-verified
//
#include <hip/hip_runtime.h>

// Problem constants (match the reference)
#define B_  16
#define SA_ 2048
#define SL_ 2048
#define DA_ 256
#define DL_ 256

typedef float v2f __attribute__((ext_vector_type(2)));
typedef float v8f __attribute__((ext_vector_type(8)));

__device__ __forceinline__ float fast_tanh(float x) {
#if __has_builtin(__builtin_amdgcn_tanhf)
    return __builtin_amdgcn_tanhf(x);   // gfx1250 v_tanh_f32 (TRANS op)
#else
    return tanhf(x);
#endif
}

#define WMMA_F32(a, bm, c) \
    __builtin_amdgcn_wmma_f32_16x16x4_f32(false, (a), false, (bm), (short)0, (c), false, false)

// ---------------------------------------------------------------------------
// Kernel 1: row-wise projections.
//   rows [0, B*SA)          : ws[row] = A[row,:] . u_w
//   rows [B*SA, B*SA+B*SL)  : ws[row] = L[row-B*SA,:] . v_w + v_b
// One wave (32 lanes) per row; coalesced loads; shuffle reduction.
// ---------------------------------------------------------------------------
__global__ __launch_bounds__(256) void didi_proj_kernel(
    const float* __restrict__ A, const float* __restrict__ L,
    const float* __restrict__ u_w, const float* __restrict__ v_w,
    const float* __restrict__ v_b, float* __restrict__ ws)
{
    const int lane = threadIdx.x & 31;
    const int wave = threadIdx.x >> 5;
    const int row  = blockIdx.x * 8 + wave;           // 8 waves per block
    const int NA   = B_ * SA_;

    float acc = 0.0f;
    if (row < NA) {
        const float* src = A + (size_t)row * DA_;
#pragma unroll
        for (int i = 0; i < DA_ / 32; ++i)
            acc += src[lane + 32 * i] * u_w[lane + 32 * i];
    } else {
        const float* src = L + (size_t)(row - NA) * DL_;
#pragma unroll
        for (int i = 0; i < DL_ / 32; ++i)
            acc += src[lane + 32 * i] * v_w[lane + 32 * i];
    }
#pragma unroll
    for (int m = 16; m >= 1; m >>= 1)
        acc += __shfl_xor(acc, m, 32);

    if (lane == 0)
        ws[row] = (row < NA) ? acc : acc + v_b[0];
}

// ---------------------------------------------------------------------------
// Kernel 2: fused masked tanh-attention + GEMM, one pass over s.
//   Block = 512 threads = 16 waves; block owns 256(t) x 64(d) of batch b.
//   Wave w -> t-tile rows [t0, t0+16), with FOUR 16x16 f32 accumulators
//   covering d columns d0+{0,16,32,48}; one score A-operand feeds all four
//   V_WMMA_F32_16X16X4_F32 per K-step (score redundancy 4x, not 16x).
//
//   K-loop is mask-free: s in [0, len_a & ~3) (exact — masked scores are 0
//   and contribute nothing), plus one masked tail step; s >= len_a skipped.
//   Fully-masked t-tiles (t0 >= len_l) write zeros and exit early.
// ---------------------------------------------------------------------------
__global__ __launch_bounds__(512) void didi_attn_kernel(
    const float* __restrict__ A,
    const int*   __restrict__ length_a, const int* __restrict__ length_l,
    const float* __restrict__ ws, float* __restrict__ out)
{
    __shared__ float ua_s[SA_];                       // 8 KB

    const int b    = blockIdx.z;
    const int lane = threadIdx.x & 31;
    const int wave = threadIdx.x >> 5;
    const int t0   = blockIdx.x * 256 + wave * 16;    // 16 waves = 16 t-tiles
    const int d0   = blockIdx.y * 64;                 // 4 d-subtiles per wave

    const int len_a = length_a[b];
    const int len_l = length_l[b];

    const float* ua_g = ws + (size_t)b * SA_;
    const float* vl_g = ws + (size_t)B_ * SA_ + (size_t)b * SL_;

    // Stage ua[b,:] into LDS once per workgroup (single barrier, then none).
    for (int i = threadIdx.x; i < SA_; i += 512)
        ua_s[i] = ua_g[i];
    __syncthreads();

    const int half = lane >> 4;                       // 0 or 1
    const int lid  = lane & 15;
    const int dcol = d0 + lid;

    // Whole 16-row t-tile masked out -> write zeros, skip the K loop.
    if (t0 >= len_l) {
#pragma unroll
        for (int v = 0; v < 8; ++v) {
            const size_t row = ((size_t)b * SL_ + (t0 + v + half * 8)) * DA_ + dcol;
            out[row]      = 0.0f;
            out[row + 16] = 0.0f;
            out[row + 32] = 0.0f;
            out[row + 48] = 0.0f;
        }
        return;
    }

    const float vl_t  = vl_g[t0 + lid];               // row value for A-operand
    const float* Abase = A + (size_t)b * SA_ * DA_ + dcol;

    v8f c0 = {0.f,0.f,0.f,0.f,0.f,0.f,0.f,0.f};
    v8f c1 = c0, c2 = c0, c3 = c0;
    float norm_acc = 0.0f;

    const int full = len_a & ~3;                      // mask-free span

#pragma unroll 4
    for (int s0 = 0; s0 < full; s0 += 4) {
        // lanes 0-15: K0/K1 = s0,s0+1 ; lanes 16-31: K2/K3 = s0+2,s0+3
        float2 u2 = reinterpret_cast<const float2*>(ua_s)[(s0 >> 1) + half];
        v2f a;
        a.x = fast_tanh(vl_t + u2.x);
        a.y = fast_tanh(vl_t + u2.y);
        norm_acc += a.x + a.y;

        const float* ap = Abase + (size_t)(s0 + half * 2) * DA_;
        v2f b0, b1, b2, b3;
        b0.x = ap[0];       b0.y = ap[DA_];
        b1.x = ap[16];      b1.y = ap[DA_ + 16];
        b2.x = ap[32];      b2.y = ap[DA_ + 32];
        b3.x = ap[48];      b3.y = ap[DA_ + 48];

        c0 = WMMA_F32(a, b0, c0);
        c1 = WMMA_F32(a, b1, c1);
        c2 = WMMA_F32(a, b2, c2);
        c3 = WMMA_F32(a, b3, c3);
    }

    // Masked tail: at most one K-step (only when len_a % 4 != 0, so
    // full <= 2044 and all LDS/global row indices stay in bounds).
    if (full < len_a) {
        const int s0 = full;
        const int s  = s0 + half * 2;
        float2 u2 = reinterpret_cast<const float2*>(ua_s)[(s0 >> 1) + half];
        v2f a;
        a.x = (s     < len_a) ? fast_tanh(vl_t + u2.x) : 0.0f;
        a.y = (s + 1 < len_a) ? fast_tanh(vl_t + u2.y) : 0.0f;
        norm_acc += a.x + a.y;

        const float* ap = Abase + (size_t)s * DA_;
        v2f b0, b1, b2, b3;
        b0.x = ap[0];       b0.y = ap[DA_];
        b1.x = ap[16];      b1.y = ap[DA_ + 16];
        b2.x = ap[32];      b2.y = ap[DA_ + 32];
        b3.x = ap[48];      b3.y = ap[DA_ + 48];

        c0 = WMMA_F32(a, b0, c0);
        c1 = WMMA_F32(a, b1, c1);
        c2 = WMMA_F32(a, b2, c2);
        c3 = WMMA_F32(a, b3, c3);
    }

    // norm[t0+j] split across lanes j and j+16 -> combine.
    float norm = norm_acc + __shfl_xor(norm_acc, 16, 32);

    // C/D layout: VGPR v holds row M = v + 8*half, col N = lid.
#pragma unroll
    for (int v = 0; v < 8; ++v) {
        const int m = v + half * 8;
        const int t = t0 + m;
        float nv    = __shfl(norm, m, 32);            // norm for row t0+m
        float scale = (t < len_l) ? (1.0f / nv) : 0.0f;   // normalize + mask_l
        const size_t row = ((size_t)b * SL_ + t) * DA_ + dcol;
        out[row]      = c0[v] * scale;
        out[row + 16] = c1[v] * scale;
        out[row + 32] = c2[v] * scale;
        out[row + 48] = c3[v] * scale;
    }
}

// ---------------------------------------------------------------------------
extern "C" void kernel_launch(void* const* d_in, const int* in_sizes, int n_in,
                              void* d_out, int out_size, void* d_ws, size_t ws_size,
                              hipStream_t stream)
{
    const float* A        = (const float*)d_in[0];
    const float* L        = (const float*)d_in[1];
    const int*   length_a = (const int*)  d_in[2];
    const int*   length_l = (const int*)  d_in[3];
    const float* u_w      = (const float*)d_in[4];
    const float* v_w      = (const float*)d_in[5];
    const float* v_b      = (const float*)d_in[6];
    float*       out      = (float*)d_out;
    float*       ws       = (float*)d_ws;   // [B*SA] ua ++ [B*SL] vl (256 KB)

    // Kernel 1: 65536 rows, 8 waves (rows) per 256-thread block.
    const int rows    = B_ * (SA_ + SL_);
    const int blocks1 = rows / 8;
    didi_proj_kernel<<<blocks1, 256, 0, stream>>>(A, L, u_w, v_w, v_b, ws);

    // Kernel 2: grid = (SL/256, DA/64, B) = (8, 4, 16), 512 threads.
    dim3 grid2(SL_ / 256, DA_ / 64, B_);
    didi_attn_kernel<<<grid2, 512, 0, stream>>>(A, length_a, length_l, ws, out);
}